// GNN_4612794876017
// MI455X (gfx1250) — compile-verified
//
#include <hip/hip_runtime.h>

// ---------------------------------------------------------------------------
// 2-layer GraphSAGE on gfx1250 (MI455X).
//   layer: agg = segment_mean(x[src] -> dst);  out = agg@W_l + x@W_r + b
// Edge phase is L2-resident scatter/gather (feature tables ~25.6MB each fit in
// the 192MB L2); dense phase uses V_WMMA_F32_16X16X4_F32 for exact-f32 GEMM.
// ---------------------------------------------------------------------------

typedef __attribute__((ext_vector_type(2))) float v2f;
typedef __attribute__((ext_vector_type(8))) float v8f;

#define N_NODES 50000
#define N_EDGES 1600000
#define IN_DIM  128
#define HIDDEN  128
#define OUT_DIM 64

// ---------------------------------------------------------------------------
// Scatter: one wave (32 lanes) per edge. Each lane moves 4 consecutive floats
// (32 lanes * 4 = 128 dims). Hardware f32 atomics (global_atomic_add_f32) at
// the L2 atomic units. Lane 0 optionally bumps the degree counter.
// ---------------------------------------------------------------------------
__global__ __launch_bounds__(256)
void sage_scatter(const float* __restrict__ feat,
                  const int*   __restrict__ src,
                  const int*   __restrict__ dst,
                  float* __restrict__ agg,
                  float* __restrict__ deg,   // may be nullptr (layer 2)
                  int n_edges) {
    const int lane = threadIdx.x & 31;
    const int e    = blockIdx.x * 8 + (threadIdx.x >> 5);
    if (e >= n_edges) return;                       // wave-uniform guard

    const int s = src[e];
    const int d = dst[e];

    const float4 v =
        *(reinterpret_cast<const float4*>(feat + (size_t)s * 128) + lane);

    float* out = agg + (size_t)d * 128 + lane * 4;
    unsafeAtomicAdd(out + 0, v.x);
    unsafeAtomicAdd(out + 1, v.y);
    unsafeAtomicAdd(out + 2, v.z);
    unsafeAtomicAdd(out + 3, v.w);

    if (deg != nullptr && lane == 0) {
        unsafeAtomicAdd(deg + d, 1.0f);
    }
}

// ---------------------------------------------------------------------------
// Fused dual-GEMM tile kernel:
//   out[16x16 tile] = relu?( (agg/deg) @ W_l + X @ W_r + bias )
// One wave per 16x16 output tile. K = 128 -> 32 k-steps, 2 WMMAs per step.
//
// V_WMMA_F32_16X16X4_F32 operand layout (ISA 7.12.2):
//   A (16x4):  lanes 0-15 hold M=lane, {K=k0,k0+1}; lanes 16-31 hold {k0+2,k0+3}
//   B (4x16):  lanes 0-15 hold N=lane, {K=k0,k0+1}; lanes 16-31 hold {k0+2,k0+3}
//   C/D:       VGPR r: lanes 0-15 -> M=r, lanes 16-31 -> M=r+8; N = lane&15
// ---------------------------------------------------------------------------
template <int NDIM, bool RELU>
__global__ __launch_bounds__(256)
void sage_gemm(const float* __restrict__ agg,
               const float* __restrict__ deg,
               const float* __restrict__ X,     // root features [n, 128]
               const float* __restrict__ Wl,    // [128, NDIM]
               const float* __restrict__ Wr,    // [128, NDIM]
               const float* __restrict__ bias,  // [NDIM]
               float* __restrict__ out,         // [n, NDIM]
               int n_nodes) {
    constexpr int NTILES_N = NDIM / 16;
    const int wave = blockIdx.x * 8 + (threadIdx.x >> 5);
    const int total_tiles = (n_nodes / 16) * NTILES_N;
    if (wave >= total_tiles) return;             // wave-uniform guard

    const int mtile = wave / NTILES_N;
    const int ntile = wave % NTILES_N;

    const int lane = threadIdx.x & 31;
    const int r    = lane & 15;                  // row within tile (A), col (B)
    const int kk   = (lane >> 4) * 2;            // K sub-offset: 0 or 2

    const int row  = mtile * 16 + r;
    const int coln = ntile * 16 + r;

    const float dval = deg[row];
    const float invd = 1.0f / fmaxf(dval, 1.0f);

    const float* aggRow = agg + (size_t)row * 128;
    const float* xRow   = X   + (size_t)row * 128;

    v8f acc = {};

#pragma unroll 8
    for (int k0 = 0; k0 < 128; k0 += 4) {
        const int k = k0 + kk;

        v2f aA; aA.x = aggRow[k] * invd;  aA.y = aggRow[k + 1] * invd;
        v2f aX; aX.x = xRow[k];           aX.y = xRow[k + 1];

        v2f bL; bL.x = Wl[(size_t)k * NDIM + coln];
                bL.y = Wl[(size_t)(k + 1) * NDIM + coln];
        v2f bR; bR.x = Wr[(size_t)k * NDIM + coln];
                bR.y = Wr[(size_t)(k + 1) * NDIM + coln];

        acc = __builtin_amdgcn_wmma_f32_16x16x4_f32(
                  false, aA, false, bL, (short)0, acc, false, false);
        acc = __builtin_amdgcn_wmma_f32_16x16x4_f32(
                  false, aX, false, bR, (short)0, acc, false, false);
    }

    // Epilogue: bias (+ReLU), store per C/D layout.
    const int   half = lane >> 4;
    const float bv   = bias[coln];
#pragma unroll
    for (int i = 0; i < 8; ++i) {
        const int m    = i + 8 * half;
        const int node = mtile * 16 + m;
        float v = acc[i] + bv;
        if (RELU) v = fmaxf(v, 0.0f);
        out[(size_t)node * NDIM + (ntile * 16 + (lane & 15))] = v;
    }
}

// ---------------------------------------------------------------------------
// Host-side launcher. Inputs (setup_inputs order):
//   0: x [50000,128] f32       1: edge_index [2,1.6M] int
//   2: W1_l [128,128] f32      3: W1_r [128,128] f32    4: b1 [128] f32
//   5: W2_l [128,64]  f32      6: W2_r [128,64]  f32    7: b2 [64]  f32
// Output: [50000, 64] f32.
// Workspace layout: agg1 | agg2 | deg | h   (agg1,agg2,deg zeroed per call)
// ---------------------------------------------------------------------------
extern "C" void kernel_launch(void* const* d_in, const int* in_sizes, int n_in,
                              void* d_out, int out_size, void* d_ws, size_t ws_size,
                              hipStream_t stream) {
    const float* x    = (const float*)d_in[0];
    const int*   ei   = (const int*)  d_in[1];
    const float* W1l  = (const float*)d_in[2];
    const float* W1r  = (const float*)d_in[3];
    const float* b1   = (const float*)d_in[4];
    const float* W2l  = (const float*)d_in[5];
    const float* W2r  = (const float*)d_in[6];
    const float* b2   = (const float*)d_in[7];
    float*       outp = (float*)d_out;

    const int n_nodes = in_sizes[0] / IN_DIM;     // 50000
    const int n_edges = in_sizes[1] / 2;          // 1600000
    const int* src = ei;
    const int* dst = ei + n_edges;

    const size_t aggBytes = (size_t)n_nodes * 128 * sizeof(float);
    const size_t degBytes = (((size_t)n_nodes * sizeof(float)) + 255) & ~(size_t)255;

    char* ws    = (char*)d_ws;
    float* agg1 = (float*)(ws);
    float* agg2 = (float*)(ws + aggBytes);
    float* deg  = (float*)(ws + 2 * aggBytes);
    float* h    = (float*)(ws + 2 * aggBytes + degBytes);

    // Zero the accumulation buffers + degrees every call (deterministic replay).
    hipMemsetAsync(d_ws, 0, 2 * aggBytes + degBytes, stream);

    const int scatterBlocks = (n_edges + 7) / 8;          // 8 edges / block
    const int tiles1 = (n_nodes / 16) * (HIDDEN / 16);    // 25000 waves
    const int tiles2 = (n_nodes / 16) * (OUT_DIM / 16);   // 12500 waves

    // Layer 1
    sage_scatter<<<scatterBlocks, 256, 0, stream>>>(x, src, dst, agg1, deg, n_edges);
    sage_gemm<HIDDEN, true><<<(tiles1 + 7) / 8, 256, 0, stream>>>(
        agg1, deg, x, W1l, W1r, b1, h, n_nodes);

    // Layer 2 (degrees unchanged — reuse)
    sage_scatter<<<scatterBlocks, 256, 0, stream>>>(h, src, dst, agg2, nullptr, n_edges);
    sage_gemm<OUT_DIM, false><<<(tiles2 + 7) / 8, 256, 0, stream>>>(
        agg2, deg, h, W2l, W2r, b2, outp, n_nodes);
}